// ArcFaceLoss_69767448756505
// MI455X (gfx1250) — compile-verified
//
#include <hip/hip_runtime.h>

// ---------------- problem constants ----------------
#define B_ROWS 512
#define DIMS 512
#define NCLS 100000
#define CHUNKS 625          // 100000 / 160
#define CHUNK_CLASSES 160   // 10 waves * 16 classes
#define ROWTILE 16
#define LDS_STRIDE 516      // 16x512 A tile padded: bank-conflict-free ds_load_b64
#define COSM 0.980066577841242f   // cos(0.2)
#define SINM 0.198669330795061f   // sin(0.2)

typedef float v2f __attribute__((ext_vector_type(2)));
typedef float v8f __attribute__((ext_vector_type(8)));

// ---------------- x row normalize: xn = x / max(||x||,1e-12) ----------------
__global__ __launch_bounds__(128) void normx_kernel(const float* __restrict__ x,
                                                    float* __restrict__ xn) {
    int row = blockIdx.x;
    const float4* xr = (const float4*)(x + (size_t)row * DIMS);
    float4 v = xr[threadIdx.x];
    float ss = v.x * v.x + v.y * v.y + v.z * v.z + v.w * v.w;
    ss += __shfl_xor(ss, 1, 32);
    ss += __shfl_xor(ss, 2, 32);
    ss += __shfl_xor(ss, 4, 32);
    ss += __shfl_xor(ss, 8, 32);
    ss += __shfl_xor(ss, 16, 32);
    __shared__ float sred[4];
    int wv = threadIdx.x >> 5, ln = threadIdx.x & 31;
    if (ln == 0) sred[wv] = ss;
    __syncthreads();
    float tot = sred[0] + sred[1] + sred[2] + sred[3];
    float inv = 1.0f / fmaxf(sqrtf(tot), 1e-12f);
    float4 o;
    o.x = v.x * inv; o.y = v.y * inv; o.z = v.z * inv; o.w = v.w * inv;
    ((float4*)(xn + (size_t)row * DIMS))[threadIdx.x] = o;
}

// ---------------- weight inverse norms: winv[c] = 1/max(||w_c||,1e-12) ------
__global__ __launch_bounds__(256) void wnorm_kernel(const float* __restrict__ w,
                                                    float* __restrict__ winv) {
    int row = blockIdx.x * 8 + (threadIdx.x >> 5);
    int ln = threadIdx.x & 31;
    const float* wr = w + (size_t)row * DIMS;
    float ss = 0.f;
    #pragma unroll
    for (int it = 0; it < 4; ++it) {
        float4 v = *(const float4*)(wr + it * 128 + ln * 4);
        ss += v.x * v.x + v.y * v.y + v.z * v.z + v.w * v.w;
    }
    ss += __shfl_xor(ss, 1, 32);
    ss += __shfl_xor(ss, 2, 32);
    ss += __shfl_xor(ss, 4, 32);
    ss += __shfl_xor(ss, 8, 32);
    ss += __shfl_xor(ss, 16, 32);
    if (ln == 0) winv[row] = 1.0f / fmaxf(sqrtf(ss), 1e-12f);
}

// ---------------- GEMM pass 1: flash-softmax partials + target logit --------
__global__ __launch_bounds__(320) void arcface_gemm_pass1(
    const float* __restrict__ xn, const float* __restrict__ wgt,
    const float* __restrict__ winv, const long long* __restrict__ y,
    const float* __restrict__ rescale,
    float* __restrict__ pmax, float* __restrict__ psum,
    float* __restrict__ ztgt) {
    __shared__ float sA[ROWTILE * LDS_STRIDE];
    __shared__ int sY[ROWTILE];
    __shared__ float sM[10][16];
    __shared__ float sS[10][16];

    const int chunk = blockIdx.x;
    const int rowBase = blockIdx.y * ROWTILE;
    const int tid = threadIdx.x;

    const float* xr = xn + (size_t)rowBase * DIMS;
    for (int idx = tid; idx < ROWTILE * DIMS; idx += 320) {
        int r = idx >> 9;
        int ccol = idx & (DIMS - 1);
        sA[r * LDS_STRIDE + ccol] = xr[idx];
    }
    if (tid < ROWTILE) sY[tid] = (int)y[rowBase + tid];
    __syncthreads();

    const int lane = tid & 31;
    const int wv = tid >> 5;         // 0..9
    const int nl = lane & 15;
    const int h = lane >> 4;
    const int c = chunk * CHUNK_CLASSES + wv * 16 + nl;

    const float* aP = &sA[nl * LDS_STRIDE + 2 * h];
    const float* bP = wgt + (size_t)c * DIMS + 2 * h;

    // two independent accumulator chains to break WMMA->WMMA RAW serialization
    v8f acc0 = {}, acc1 = {};
    #pragma unroll 4
    for (int kb = 0; kb < DIMS; kb += 8) {
        if ((kb & 31) == 0) __builtin_prefetch(bP + kb + 256, 0, 1);
        v2f a0 = *(const v2f*)(aP + kb);
        v2f b0 = *(const v2f*)(bP + kb);
        v2f a1 = *(const v2f*)(aP + kb + 4);
        v2f b1 = *(const v2f*)(bP + kb + 4);
        acc0 = __builtin_amdgcn_wmma_f32_16x16x4_f32(false, a0, false, b0,
                                                     (short)0, acc0, false, false);
        acc1 = __builtin_amdgcn_wmma_f32_16x16x4_f32(false, a1, false, b1,
                                                     (short)0, acc1, false, false);
    }
    v8f acc = acc0 + acc1;

    const float rs = rescale[0];
    const float wi = winv[c];

    float zm[8], zp[8];
    #pragma unroll
    for (int i = 0; i < 8; ++i) {
        float t = acc[i] * wi;
        t = fminf(fmaxf(t, -1.0f + 1e-8f), 1.0f - 1e-8f);
        int row = i + 8 * h;
        float zc = t;
        bool istgt = (sY[row] == c);
        if (istgt) zc = t * COSM - sqrtf(fmaxf(1.0f - t * t, 0.0f)) * SINM;
        float z = rs * zc;
        if (istgt) ztgt[rowBase + row] = z;   // unique writer per row
        zm[i] = z;
        zp[i] = z;
    }
    // per-row reduce over the 16 lanes of each half-wave
    #pragma unroll
    for (int i = 0; i < 8; ++i) {
        float mv = zm[i];
        mv = fmaxf(mv, __shfl_xor(mv, 1, 32));
        mv = fmaxf(mv, __shfl_xor(mv, 2, 32));
        mv = fmaxf(mv, __shfl_xor(mv, 4, 32));
        mv = fmaxf(mv, __shfl_xor(mv, 8, 32));
        float pv = expf(zp[i] - mv);
        pv += __shfl_xor(pv, 1, 32);
        pv += __shfl_xor(pv, 2, 32);
        pv += __shfl_xor(pv, 4, 32);
        pv += __shfl_xor(pv, 8, 32);
        zm[i] = mv;
        zp[i] = pv;
    }
    if (nl == 0) {
        #pragma unroll
        for (int i = 0; i < 8; ++i) {
            sM[wv][i + 8 * h] = zm[i];
            sS[wv][i + 8 * h] = zp[i];
        }
    }
    __syncthreads();
    if (tid < 16) {
        float mx = -1e30f;
        #pragma unroll
        for (int w2 = 0; w2 < 10; ++w2) mx = fmaxf(mx, sM[w2][tid]);
        float s = 0.f;
        #pragma unroll
        for (int w2 = 0; w2 < 10; ++w2) s += sS[w2][tid] * expf(sM[w2][tid] - mx);
        pmax[(size_t)chunk * B_ROWS + rowBase + tid] = mx;
        psum[(size_t)chunk * B_ROWS + rowBase + tid] = s;
    }
}

// ---------------- combine: per-row logsumexp L, target prob -----------------
__global__ __launch_bounds__(256) void combine_kernel(
    const float* __restrict__ pmax, const float* __restrict__ psum,
    const float* __restrict__ ztgt, float* __restrict__ Lrow,
    float* __restrict__ resT) {
    int r = blockIdx.x * 256 + threadIdx.x;
    if (r >= B_ROWS) return;
    float mx = -1e30f, s = 0.f;
    for (int k = 0; k < CHUNKS; ++k) {
        float m2 = pmax[(size_t)k * B_ROWS + r];
        float s2 = psum[(size_t)k * B_ROWS + r];
        float M = fmaxf(mx, m2);
        s = s * expf(mx - M) + s2 * expf(m2 - M);
        mx = M;
    }
    float L = mx + logf(s);
    Lrow[r] = L;
    resT[r] = expf(ztgt[r] - L);   // softmax prob at target class
}

// ---------------- GEMM pass 2: partial sums of exp(softmax probs) -----------
__global__ __launch_bounds__(320) void arcface_gemm_pass2(
    const float* __restrict__ xn, const float* __restrict__ wgt,
    const float* __restrict__ winv, const long long* __restrict__ y,
    const float* __restrict__ rescale, const float* __restrict__ Lrow,
    float* __restrict__ pT) {
    __shared__ float sA[ROWTILE * LDS_STRIDE];
    __shared__ int sY[ROWTILE];
    __shared__ float sL[ROWTILE];
    __shared__ float sS[10][16];

    const int chunk = blockIdx.x;
    const int rowBase = blockIdx.y * ROWTILE;
    const int tid = threadIdx.x;

    const float* xr = xn + (size_t)rowBase * DIMS;
    for (int idx = tid; idx < ROWTILE * DIMS; idx += 320) {
        int r = idx >> 9;
        int ccol = idx & (DIMS - 1);
        sA[r * LDS_STRIDE + ccol] = xr[idx];
    }
    if (tid < ROWTILE) {
        sY[tid] = (int)y[rowBase + tid];
        sL[tid] = Lrow[rowBase + tid];
    }
    __syncthreads();

    const int lane = tid & 31;
    const int wv = tid >> 5;
    const int nl = lane & 15;
    const int h = lane >> 4;
    const int c = chunk * CHUNK_CLASSES + wv * 16 + nl;

    const float* aP = &sA[nl * LDS_STRIDE + 2 * h];
    const float* bP = wgt + (size_t)c * DIMS + 2 * h;

    v8f acc0 = {}, acc1 = {};
    #pragma unroll 4
    for (int kb = 0; kb < DIMS; kb += 8) {
        if ((kb & 31) == 0) __builtin_prefetch(bP + kb + 256, 0, 1);
        v2f a0 = *(const v2f*)(aP + kb);
        v2f b0 = *(const v2f*)(bP + kb);
        v2f a1 = *(const v2f*)(aP + kb + 4);
        v2f b1 = *(const v2f*)(bP + kb + 4);
        acc0 = __builtin_amdgcn_wmma_f32_16x16x4_f32(false, a0, false, b0,
                                                     (short)0, acc0, false, false);
        acc1 = __builtin_amdgcn_wmma_f32_16x16x4_f32(false, a1, false, b1,
                                                     (short)0, acc1, false, false);
    }
    v8f acc = acc0 + acc1;

    const float rs = rescale[0];
    const float wi = winv[c];

    float tv[8];
    #pragma unroll
    for (int i = 0; i < 8; ++i) {
        float t = acc[i] * wi;
        t = fminf(fmaxf(t, -1.0f + 1e-8f), 1.0f - 1e-8f);
        int row = i + 8 * h;
        float zc = t;
        if (sY[row] == c) zc = t * COSM - sqrtf(fmaxf(1.0f - t * t, 0.0f)) * SINM;
        float z = rs * zc;
        float res = expf(z - sL[row]);   // softmax prob
        tv[i] = expf(res);               // term of second softmax's denom
    }
    #pragma unroll
    for (int i = 0; i < 8; ++i) {
        float pv = tv[i];
        pv += __shfl_xor(pv, 1, 32);
        pv += __shfl_xor(pv, 2, 32);
        pv += __shfl_xor(pv, 4, 32);
        pv += __shfl_xor(pv, 8, 32);
        tv[i] = pv;
    }
    if (nl == 0) {
        #pragma unroll
        for (int i = 0; i < 8; ++i) sS[wv][i + 8 * h] = tv[i];
    }
    __syncthreads();
    if (tid < 16) {
        float s = 0.f;
        #pragma unroll
        for (int w2 = 0; w2 < 10; ++w2) s += sS[w2][tid];
        pT[(size_t)chunk * B_ROWS + rowBase + tid] = s;
    }
}

// ---------------- final: nll = log T - res_y ; mean over rows ---------------
__global__ __launch_bounds__(512) void final_kernel(const float* __restrict__ pT,
                                                    const float* __restrict__ resT,
                                                    float* __restrict__ out) {
    __shared__ float red[512];
    int r = threadIdx.x;
    float T = 0.f;
    for (int k = 0; k < CHUNKS; ++k) T += pT[(size_t)k * B_ROWS + r];
    red[r] = logf(T) - resT[r];
    __syncthreads();
    for (int st = 256; st > 0; st >>= 1) {
        if (r < st) red[r] += red[r + st];
        __syncthreads();
    }
    if (r == 0) out[0] = red[0] / (float)B_ROWS;
}

extern "C" void kernel_launch(void* const* d_in, const int* in_sizes, int n_in,
                              void* d_out, int out_size, void* d_ws, size_t ws_size,
                              hipStream_t stream) {
    (void)in_sizes; (void)n_in; (void)out_size; (void)ws_size;
    const float* x = (const float*)d_in[0];
    const long long* y = (const long long*)d_in[1];
    const float* w = (const float*)d_in[2];
    const float* rescale = (const float*)d_in[3];
    float* out = (float*)d_out;
    float* ws = (float*)d_ws;

    float* xn   = ws + 0;        // 262144 floats
    float* winv = ws + 262144;   // 100000
    float* pmax = ws + 362144;   // 320000
    float* psum = ws + 682144;   // 320000
    float* pT   = ws + 1002144;  // 320000
    float* zt   = ws + 1322144;  // 512
    float* Lr   = ws + 1322656;  // 512
    float* rT   = ws + 1323168;  // 512

    normx_kernel<<<dim3(B_ROWS), dim3(128), 0, stream>>>(x, xn);
    wnorm_kernel<<<dim3(NCLS / 8), dim3(256), 0, stream>>>(w, winv);
    arcface_gemm_pass1<<<dim3(CHUNKS, B_ROWS / ROWTILE), dim3(320), 0, stream>>>(
        xn, w, winv, y, rescale, pmax, psum, zt);
    combine_kernel<<<dim3(2), dim3(256), 0, stream>>>(pmax, psum, zt, Lr, rT);
    arcface_gemm_pass2<<<dim3(CHUNKS, B_ROWS / ROWTILE), dim3(320), 0, stream>>>(
        xn, w, winv, y, rescale, Lr, pT);
    final_kernel<<<dim3(1), dim3(512), 0, stream>>>(pT, rT, out);
}